// GCN_MLP_7997229105506
// MI455X (gfx1250) — compile-verified
//
#include <hip/hip_runtime.h>
#include <hip/hip_bf16.h>

// ---------------------------------------------------------------------------
// GCN (3x GCNConv) + mean-pool + 4-layer MLP for MI455X (gfx1250, wave32).
// fp32 throughout (workload is HBM/atomic bound; ~2.7 GB traffic vs ~2 GFLOP),
// all GEMMs on V_WMMA_F32_16X16X4_F32.
// ---------------------------------------------------------------------------

typedef __attribute__((ext_vector_type(2))) float v2f;
typedef __attribute__((ext_vector_type(8))) float v8f;

#define HID 32
#define NGRAPH 64

// ---------------------------------------------------------------- utilities
__global__ void zero_f32(float* __restrict__ p, size_t n) {
  size_t i = (size_t)blockIdx.x * blockDim.x + threadIdx.x;
  if (i < n) p[i] = 0.0f;
}

// deg[dst] += edge_weight  (self-loop +1 folded into compute_dinv)
__global__ void edge_deg(const int* __restrict__ dst, const float* __restrict__ ew,
                         float* __restrict__ deg, int E) {
  int e = blockIdx.x * blockDim.x + threadIdx.x;
  if (e < E) {
    __builtin_prefetch(dst + e + 4096, 0, 1);
    atomicAdd(&deg[dst[e]], ew[e]);
  }
}

// in-place: deg -> dinv = rsqrt(deg_edges + 1);  deg+1 >= 1 so guard vanishes
__global__ void compute_dinv(float* __restrict__ deg, int N) {
  int n = blockIdx.x * blockDim.x + threadIdx.x;
  if (n < N) deg[n] = rsqrtf(deg[n] + 1.0f);
}

// ------------------------------------------------------- WMMA fp32 GEMM tile
// out[rows, ncols] = A[rows, K] @ W[K, ncols]  (+bias, optional relu)
// One 16x16 output tile per wave; 4 waves (4 row-tiles) per 128-thread block.
// W column tile staged through LDS. K is a template param -> fully unrolled
// chain of v_wmma_f32_16x16x4_f32.
template <int K>
__global__ void wmma_gemm(const float* __restrict__ A, const float* __restrict__ W,
                          const float* __restrict__ bias, float* __restrict__ out,
                          int rows, int ncols, int relu) {
  __shared__ float lw[K * 16];  // [K][16] column tile of W

  const int lane = threadIdx.x & 31;
  const int waveId = threadIdx.x >> 5;
  const int colBase = blockIdx.y * 16;

  // cooperative stage of W tile (guard ragged last column tile, e.g. C=10)
  for (int i = threadIdx.x; i < K * 16; i += blockDim.x) {
    int kk = i >> 4, nn = i & 15;
    int col = colBase + nn;
    lw[i] = (col < ncols) ? W[(size_t)kk * ncols + col] : 0.0f;
  }
  __syncthreads();

  const int rowTile = blockIdx.x * 4 + waveId;
  if (rowTile * 16 >= rows) return;
  const int rowBase = rowTile * 16;

  // A 16x4 f32 layout: lanes 0-15 -> M=lane, K=k+0,k+1 ; lanes 16-31 -> K=k+2,k+3
  const int m = lane & 15;
  const int khalf = (lane >> 4) << 1;
  const float* arow = A + (size_t)(rowBase + m) * K;

  v8f c = {};
#pragma unroll
  for (int k = 0; k < K; k += 4) {
    v2f a, b;
    a.x = arow[k + khalf];
    a.y = arow[k + khalf + 1];
    // B 4x16 layout mirrors A: lane&15 -> N, lane>>4 selects K pair
    b.x = lw[(k + khalf) * 16 + m];
    b.y = lw[(k + khalf + 1) * 16 + m];
    c = __builtin_amdgcn_wmma_f32_16x16x4_f32(false, a, false, b, (short)0, c,
                                              false, false);
  }

  // C/D 16x16 f32 layout: VGPR v -> M = v + (lane>=16 ? 8 : 0), N = lane&15
  const int mOff = (lane >> 4) * 8;
  const int col = colBase + m;
  if (col < ncols) {
    float bv = bias ? bias[col] : 0.0f;
#pragma unroll
    for (int v = 0; v < 8; ++v) {
      float val = c[v] + bv;
      if (relu) val = fmaxf(val, 0.0f);
      out[(size_t)(rowBase + mOff + v) * ncols + col] = val;
    }
  }
}

// ------------------------------------------------ edge scatter (segment_sum)
// one edge per wave; 32 lanes = 32 features (H == warpSize == 32)
__global__ void gcn_scatter(const int* __restrict__ src, const int* __restrict__ dst,
                            const float* __restrict__ ew, const float* __restrict__ dinv,
                            const float* __restrict__ h, float* __restrict__ agg, int E) {
  int e = blockIdx.x * (blockDim.x >> 5) + (threadIdx.x >> 5);
  if (e >= E) return;
  int f = threadIdx.x & 31;
  int s = src[e], d = dst[e];
  __builtin_prefetch(src + e + 2048, 0, 1);
  __builtin_prefetch(dst + e + 2048, 0, 1);
  float coef = dinv[s] * ew[e] * dinv[d];
  atomicAdd(&agg[(size_t)d * HID + f], coef * h[(size_t)s * HID + f]);
}

// agg = [relu](agg + dinv^2 * lin + bias)   (self-loop term + bias epilogue)
__global__ void finish_conv(float* __restrict__ agg, const float* __restrict__ lin,
                            const float* __restrict__ dinv, const float* __restrict__ bias,
                            int N, int relu) {
  size_t i = (size_t)blockIdx.x * blockDim.x + threadIdx.x;
  if (i >= (size_t)N * HID) return;
  int n = (int)(i >> 5), f = (int)(i & 31);
  float dv = dinv[n];
  float v = agg[i] + dv * dv * lin[i] + bias[f];
  if (relu) v = fmaxf(v, 0.0f);
  agg[i] = v;
}

// --------------------------------------------------------------- mean pool
__global__ void pool_sum(const float* __restrict__ h, const int* __restrict__ batch,
                         float* __restrict__ gsum, float* __restrict__ gcnt, int N) {
  int n = blockIdx.x * (blockDim.x >> 5) + (threadIdx.x >> 5);
  if (n >= N) return;
  int f = threadIdx.x & 31;
  int b = batch[n];
  atomicAdd(&gsum[b * HID + f], h[(size_t)n * HID + f]);
  if (f == 0) atomicAdd(&gcnt[b], 1.0f);
}

__global__ void pool_div(const float* __restrict__ gsum, const float* __restrict__ gcnt,
                         float* __restrict__ g) {
  int i = blockIdx.x * blockDim.x + threadIdx.x;
  if (i < NGRAPH * HID) g[i] = gsum[i] / fmaxf(gcnt[i >> 5], 1.0f);
}

// ---------------------------------------------------------------------------
extern "C" void kernel_launch(void* const* d_in, const int* in_sizes, int n_in,
                              void* d_out, int out_size, void* d_ws, size_t ws_size,
                              hipStream_t stream) {
  const float* x   = (const float*)d_in[0];
  const float* ew  = (const float*)d_in[1];
  const float* W1  = (const float*)d_in[2];
  const float* b1  = (const float*)d_in[3];
  const float* W2  = (const float*)d_in[4];
  const float* b2  = (const float*)d_in[5];
  const float* W3  = (const float*)d_in[6];
  const float* b3  = (const float*)d_in[7];
  const float* L1w = (const float*)d_in[8];
  const float* L1b = (const float*)d_in[9];
  const float* L2w = (const float*)d_in[10];
  const float* L2b = (const float*)d_in[11];
  const float* L3w = (const float*)d_in[12];
  const float* L3b = (const float*)d_in[13];
  const float* L4w = (const float*)d_in[14];
  const float* L4b = (const float*)d_in[15];
  const int*   ei  = (const int*)d_in[16];
  const int*   bat = (const int*)d_in[17];

  const int E = in_sizes[1];
  const int N = in_sizes[17];
  const int* srcIdx = ei;       // edge_index[0, :]
  const int* dstIdx = ei + E;   // edge_index[1, :]

  // -------- workspace carve-up (fp32, 64-float aligned) --------
  float* ws = (float*)d_ws;
  size_t off = 0;
  auto carve = [&](size_t n) { float* p = ws + off; off += (n + 63) & ~(size_t)63; return p; };
  float* deg  = carve(N);                     // degree -> dinv (in place)
  float* bufL = carve((size_t)N * HID);       // linear output h @ W
  float* bufA = carve((size_t)N * HID);       // aggregation target / layer out
  float* gsum = carve(NGRAPH * HID + NGRAPH); // graph sums, then counts
  float* gcnt = gsum + NGRAPH * HID;
  float* g    = carve(NGRAPH * HID);          // pooled features
  float* m1   = carve(NGRAPH * 64);
  float* m2   = carve(NGRAPH * 64);
  (void)ws_size; (void)n_in; (void)out_size;

  const int T = 256;
  auto blocks = [&](size_t n) { return (unsigned)((n + T - 1) / T); };

  // -------- GCN normalization --------
  zero_f32<<<blocks(N), T, 0, stream>>>(deg, (size_t)N);
  edge_deg<<<blocks(E), T, 0, stream>>>(dstIdx, ew, deg, E);
  compute_dinv<<<blocks(N), T, 0, stream>>>(deg, N);  // deg now holds dinv

  auto gemm32  = [&](const float* A, const float* W, const float* bias, float* out,
                     int rows, int ncols, int relu) {
    dim3 grid(((rows + 15) / 16 + 3) / 4, (ncols + 15) / 16);
    wmma_gemm<32><<<grid, 128, 0, stream>>>(A, W, bias, out, rows, ncols, relu);
  };
  auto gemm64  = [&](const float* A, const float* W, const float* bias, float* out,
                     int rows, int ncols, int relu) {
    dim3 grid(((rows + 15) / 16 + 3) / 4, (ncols + 15) / 16);
    wmma_gemm<64><<<grid, 128, 0, stream>>>(A, W, bias, out, rows, ncols, relu);
  };

  const unsigned edgeBlocks = (unsigned)((E + 7) / 8);      // 8 edges (waves) / block
  const unsigned nodeBlocks = (unsigned)((N + 7) / 8);
  const size_t   NH = (size_t)N * HID;

  // -------- conv1: x[N,128] @ W1 -> scatter -> +selfloop +b1, relu --------
  {
    dim3 grid(((N + 15) / 16 + 3) / 4, HID / 16);
    wmma_gemm<128><<<grid, 128, 0, stream>>>(x, W1, nullptr, bufL, N, HID, 0);
  }
  zero_f32<<<blocks(NH), T, 0, stream>>>(bufA, NH);
  gcn_scatter<<<edgeBlocks, T, 0, stream>>>(srcIdx, dstIdx, ew, deg, bufL, bufA, E);
  finish_conv<<<blocks(NH), T, 0, stream>>>(bufA, bufL, deg, b1, N, 1);

  // -------- conv2 --------
  gemm32(bufA, W2, nullptr, bufL, N, HID, 0);
  zero_f32<<<blocks(NH), T, 0, stream>>>(bufA, NH);
  gcn_scatter<<<edgeBlocks, T, 0, stream>>>(srcIdx, dstIdx, ew, deg, bufL, bufA, E);
  finish_conv<<<blocks(NH), T, 0, stream>>>(bufA, bufL, deg, b2, N, 1);

  // -------- conv3 (no relu) --------
  gemm32(bufA, W3, nullptr, bufL, N, HID, 0);
  zero_f32<<<blocks(NH), T, 0, stream>>>(bufA, NH);
  gcn_scatter<<<edgeBlocks, T, 0, stream>>>(srcIdx, dstIdx, ew, deg, bufL, bufA, E);
  finish_conv<<<blocks(NH), T, 0, stream>>>(bufA, bufL, deg, b3, N, 0);

  // -------- global mean pool --------
  zero_f32<<<blocks(NGRAPH * HID + NGRAPH), T, 0, stream>>>(gsum,
                                                            (size_t)(NGRAPH * HID + NGRAPH));
  pool_sum<<<nodeBlocks, T, 0, stream>>>(bufA, bat, gsum, gcnt, N);
  pool_div<<<blocks(NGRAPH * HID), T, 0, stream>>>(gsum, gcnt, g);

  // -------- MLP head: 32 -> 64 -> 64 -> 64 -> 10 --------
  gemm32(g,  L1w, L1b, m1, NGRAPH, 64, 1);
  gemm64(m1, L2w, L2b, m2, NGRAPH, 64, 1);
  gemm64(m2, L3w, L3b, m1, NGRAPH, 64, 1);
  gemm64(m1, L4w, L4b, (float*)d_out, NGRAPH, 10, 0);
}